// NormalizedCrossCorrelation_47150150975542
// MI455X (gfx1250) — compile-verified
//
#include <hip/hip_runtime.h>

// NCC loss: per-column reductions (Parseval eliminates the FFT), memory-bound
// streaming kernel + deterministic tree reduction finished with a f32 WMMA.

typedef float v2f __attribute__((ext_vector_type(2)));
typedef float v8f __attribute__((ext_vector_type(8)));

#define NT    4096
#define NCOL  3072              // ntr * nc = 1024*3
#define NQ    (NCOL / 4)        // 768 float4 column-quads
#define K1_BLOCK 128
#define K1_CB    (NQ / K1_BLOCK) // 6 column-blocks
#define NCH   64                // t-chunks
#define TCH   (NT / NCH)        // 64 t per chunk
#define K2_BLOCK 64
#define K2_NB    (NCOL / K2_BLOCK) // 48
#define DT_C  0.001f
#define EPS_C 1e-10f

// ---------------- Kernel 1: streaming partial reductions ----------------
// grid (6, 64), block 128. Thread owns 4 consecutive columns (one float4),
// reduces TCH time samples. Coalesced float4 loads: addr = t*NQ + q.
__global__ void ncc_partial(const float* __restrict__ xg,
                            const float* __restrict__ yg,
                            float* __restrict__ ws) {
    const int q  = blockIdx.x * K1_BLOCK + threadIdx.x;   // 0..NQ-1
    const int t0 = blockIdx.y * TCH;
    const float4* __restrict__ x4 = (const float4*)xg;
    const float4* __restrict__ y4 = (const float4*)yg;

    float4 sxy = {0.f,0.f,0.f,0.f};
    float4 sxx = {0.f,0.f,0.f,0.f};
    float4 syy = {0.f,0.f,0.f,0.f};
    float4 mx  = {0.f,0.f,0.f,0.f};

#pragma unroll 4
    for (int t = t0; t < t0 + TCH; ++t) {
        float4 a = x4[t * NQ + q];
        float4 b = y4[t * NQ + q];
        sxy.x = fmaf(a.x, b.x, sxy.x); sxy.y = fmaf(a.y, b.y, sxy.y);
        sxy.z = fmaf(a.z, b.z, sxy.z); sxy.w = fmaf(a.w, b.w, sxy.w);
        sxx.x = fmaf(a.x, a.x, sxx.x); sxx.y = fmaf(a.y, a.y, sxx.y);
        sxx.z = fmaf(a.z, a.z, sxx.z); sxx.w = fmaf(a.w, a.w, sxx.w);
        syy.x = fmaf(b.x, b.x, syy.x); syy.y = fmaf(b.y, b.y, syy.y);
        syy.z = fmaf(b.z, b.z, syy.z); syy.w = fmaf(b.w, b.w, syy.w);
        mx.x = fmaxf(mx.x, fabsf(a.x)); mx.y = fmaxf(mx.y, fabsf(a.y));
        mx.z = fmaxf(mx.z, fabsf(a.z)); mx.w = fmaxf(mx.w, fabsf(a.w));
    }

    // ws layout (floats): [chunk][quantity(4)][NCOL]; write as float4 per quad
    float4* w = (float4*)ws;
    const int base = blockIdx.y * 4 * NQ;
    w[base + 0 * NQ + q] = sxy;
    w[base + 1 * NQ + q] = sxx;
    w[base + 2 * NQ + q] = syy;
    w[base + 3 * NQ + q] = mx;
}

// ---------------- Kernel 2: per-column cc + WMMA block reduction ----------------
// grid 48, block 64. Thread owns one column: fold NCH chunk partials, compute
// masked cc, then reduce 64 cc values with one V_WMMA_F32_16X16X4_F32
// (B = ones -> D[m][n] = rowsum(A[m]), exact f32).
__global__ void ncc_columns(const float* __restrict__ ws,
                            float* __restrict__ part) {
    __shared__ float red[K2_BLOCK];
    const int c = blockIdx.x * K2_BLOCK + threadIdx.x;    // 0..NCOL-1

    float sxy = 0.f, sxx = 0.f, syy = 0.f, mx = 0.f;
    for (int ch = 0; ch < NCH; ++ch) {
        const float* b = ws + ch * 4 * NCOL;
        sxy += b[0 * NCOL + c];
        sxx += b[1 * NCOL + c];
        syy += b[2 * NCOL + c];
        mx   = fmaxf(mx, b[3 * NCOL + c]);
    }
    const float Ex = fmaf(DT_C, sxx, EPS_C);
    const float Ey = fmaf(DT_C, syy, EPS_C);
    const float cc = (mx > 0.0f) ? (sxy / sqrtf(Ex * Ey)) : 0.0f;

    red[threadIdx.x] = cc;
    __syncthreads();

    if (threadIdx.x < 32) {   // whole wave 0 active -> EXEC all ones for WMMA
        const int lane = threadIdx.x;
        // 32-bit A 16x4 layout: lanes 0-15 hold K=0,1 (VGPR0,1); lanes 16-31 K=2,3
        v2f a;
        if (lane < 16) { a.x = red[lane * 4 + 0];        a.y = red[lane * 4 + 1]; }
        else           { a.x = red[(lane - 16) * 4 + 2]; a.y = red[(lane - 16) * 4 + 3]; }
        v2f bones = {1.0f, 1.0f};   // all-ones B: layout-independent
        v8f cacc = {};
        cacc = __builtin_amdgcn_wmma_f32_16x16x4_f32(
            /*neg_a=*/false, a, /*neg_b=*/false, bones,
            /*c_mod=*/(short)0, cacc, /*reuse_a=*/false, /*reuse_b=*/false);
        // D[m][n] = rowsum[m]; lanes 0-15 hold rows 0-7 in d[0..7], lanes 16-31 rows 8-15
        float s = cacc[0] + cacc[1] + cacc[2] + cacc[3] +
                  cacc[4] + cacc[5] + cacc[6] + cacc[7];
        float hi = __shfl(s, 16, 32);
        if (lane == 0) part[blockIdx.x] = s + hi;
    }
}

// ---------------- Kernel 3: final scalar ----------------
__global__ void ncc_final(const float* __restrict__ part, float* __restrict__ out) {
    if (threadIdx.x == 0) {
        float s = 0.f;
        for (int i = 0; i < K2_NB; ++i) s += part[i];
        out[0] = s;
    }
}

extern "C" void kernel_launch(void* const* d_in, const int* in_sizes, int n_in,
                              void* d_out, int out_size, void* d_ws, size_t ws_size,
                              hipStream_t stream) {
    const float* x = (const float*)d_in[0];
    const float* y = (const float*)d_in[1];
    float* ws   = (float*)d_ws;                 // NCH*4*NCOL floats (~3 MB)
    float* part = ws + (size_t)NCH * 4 * NCOL;  // + K2_NB floats
    (void)in_sizes; (void)n_in; (void)out_size; (void)ws_size;

    ncc_partial<<<dim3(K1_CB, NCH), K1_BLOCK, 0, stream>>>(x, y, ws);
    ncc_columns<<<K2_NB, K2_BLOCK, 0, stream>>>(ws, part);
    ncc_final<<<1, 32, 0, stream>>>(part, (float*)d_out);
}